// LBP_39779987096284
// MI455X (gfx1250) — compile-verified
//
#include <hip/hip_runtime.h>
#include <cstdint>

// LBP forward for MI455X (gfx1250, wave32).
// Shapes: in (32,64,56,56) f32; kernels (128,4,2) i32; pmap (128,4) i32;
// out (32,128,56,56) f32.

#define N_      32
#define D_      64
#define H_      56
#define W_      56
#define F_      128
#define ROWS    4                       // output rows per block
#define LROWS   (ROWS + 2)              // 6 rows incl. halo
#define LSTRIDE 64                      // padded floats per LDS row (data at cols 4..59)
#define CSTRIDE (LROWS * LSTRIDE)       // 384 floats per channel
#define THREADS (ROWS * W_)             // 224 = 7 waves of 32
#define NROWBLK (H_ / ROWS)             // 14
#define CHUNKS  (W_ / 4)                // 14 b128 chunks per row
#define FSPLIT  2                       // filter-range split across gridDim.y
#define FPB     (F_ / FSPLIT)           // 64 filters per block

#ifndef __has_builtin
#define __has_builtin(x) 0
#endif

#if __has_builtin(__builtin_amdgcn_tanhf)
#define TANHF(x) __builtin_amdgcn_tanhf(x)
#else
__device__ __forceinline__ float tanh_fallback(float x) {
    // tanh(x) = (1 - e)/(1 + e), e = exp(-2x) = exp2(-2*log2e*x)
    float e = __builtin_amdgcn_exp2f(-2.885390082f * x);
    return (1.0f - e) * __builtin_amdgcn_rcpf(1.0f + e);
}
#define TANHF(x) tanh_fallback(x)
#endif

__global__ __launch_bounds__(THREADS)
void lbp_forward_kernel(const float* __restrict__ in,
                        const int*   __restrict__ kers,
                        const int*   __restrict__ pmap,
                        float*       __restrict__ out) {
    // 64 ch * 6 rows * 64 floats * 4B = 96 KB (of 320 KB per WGP)
    __shared__ float4 lds4[(D_ * CSTRIDE) / 4];
    float* lds = (float*)lds4;

    const int tid = threadIdx.x;
    const int n   = blockIdx.x / NROWBLK;
    const int h0  = (blockIdx.x % NROWBLK) * ROWS;
    const int f0  = blockIdx.y * FPB;

    // ---- 1) zero-fill LDS (realizes the spatial zero padding) ----
    const float4 z = make_float4(0.f, 0.f, 0.f, 0.f);
    for (int i = tid; i < (D_ * CSTRIDE) / 4; i += THREADS)
        lds4[i] = z;
    __syncthreads();

    // ---- 2) async-copy all 64 channels of the 6-row strip into LDS ----
    // Each task moves 16B (4 floats). Data placed at col offset 4 -> 16B aligned.
    const float* gin = in + (size_t)n * (D_ * H_ * W_);
    for (int t = tid; t < D_ * LROWS * CHUNKS; t += THREADS) {
        int c  = t / (LROWS * CHUNKS);
        int r  = (t / CHUNKS) % LROWS;
        int ch = t % CHUNKS;
        int gr = h0 - 1 + r;
        if ((unsigned)gr < (unsigned)H_) {
            uint32_t loff =
                (uint32_t)(uintptr_t)&lds[(c * LROWS + r) * LSTRIDE + 4 + ch * 4];
            uint64_t gaddr = (uint64_t)(uintptr_t)
                (gin + ((size_t)c * H_ + gr) * W_ + ch * 4);
            asm volatile("global_load_async_to_lds_b128 %0, %1, off"
                         :: "v"(loff), "v"(gaddr) : "memory");
        }
    }
    asm volatile("s_wait_asynccnt 0x0" ::: "memory");
    __syncthreads();

    // ---- 3) compute: 64 filters per block, 1 pixel per thread ----
    const int lh   = tid / W_;             // 0..3
    const int w    = tid - lh * W_;        // 0..55
    const int rowA = lh << 6;              // lh * LSTRIDE
    const int colB = 3 + w;                // col for kx offset (kx in 0..2 -> -1..+1)
    const int cenB = ((lh + 1) << 6) + 4 + w;

    float* outn = out + (size_t)n * (F_ * H_ * W_)
                      + (size_t)f0 * (H_ * W_)
                      + (size_t)(h0 + lh) * W_ + w;

    #pragma unroll 2
    for (int fi = 0; fi < FPB; ++fi) {
        const int f = f0 + fi;
        // Uniform in f -> scalar (s_load) descriptor fetches
        const int4 pm = *(const int4*)(pmap + (f << 2));     // channels p=0..3
        const int4 ka = *(const int4*)(kers + (f << 3));     // ky0,kx0,ky1,kx1
        const int4 kb = *(const int4*)(kers + (f << 3) + 4); // ky2,kx2,ky3,kx3

        const int b0 = pm.x * CSTRIDE;
        const int b1 = pm.y * CSTRIDE;
        const int b2 = pm.z * CSTRIDE;
        const int b3 = pm.w * CSTRIDE;

        float s0 = lds[b0 + rowA + (ka.x << 6) + colB + ka.y];
        float c0 = lds[b0 + cenB];
        float s1 = lds[b1 + rowA + (ka.z << 6) + colB + ka.w];
        float c1 = lds[b1 + cenB];
        float s2 = lds[b2 + rowA + (kb.x << 6) + colB + kb.y];
        float c2 = lds[b2 + cenB];
        float s3 = lds[b3 + rowA + (kb.z << 6) + colB + kb.w];
        float c3 = lds[b3 + cenB];

        // sum_p 2^p * sigmoid((s_p - c_p)/0.1) == 7.5 + sum_p 2^(p-1)*tanh(5*(s_p-c_p))
        float acc = 7.5f;
        acc += 0.5f * TANHF(5.0f * (s0 - c0));
        acc += 1.0f * TANHF(5.0f * (s1 - c1));
        acc += 2.0f * TANHF(5.0f * (s2 - c2));
        acc += 4.0f * TANHF(5.0f * (s3 - c3));

        // Output is write-once, never re-read: non-temporal (TH=NT) keeps the
        // L2-resident input + descriptor tables hot.
        __builtin_nontemporal_store(acc, outn + (size_t)fi * (H_ * W_));
    }
}

extern "C" void kernel_launch(void* const* d_in, const int* in_sizes, int n_in,
                              void* d_out, int out_size, void* d_ws, size_t ws_size,
                              hipStream_t stream) {
    const float* in   = (const float*)d_in[0];
    const int*   kers = (const int*)d_in[1];
    const int*   pmap = (const int*)d_in[2];
    float*       out  = (float*)d_out;

    dim3 grid(N_ * NROWBLK, FSPLIT);   // 448 x 2 blocks
    dim3 block(THREADS);               // 224 threads (7 wave32)
    lbp_forward_kernel<<<grid, block, 0, stream>>>(in, kers, pmap, out);
}